// LocalEncoder_69844758167736
// MI455X (gfx1250) — compile-verified
//
#include <hip/hip_runtime.h>
#include <math.h>

typedef float v2f __attribute__((ext_vector_type(2)));
typedef float v8f __attribute__((ext_vector_type(8)));
typedef unsigned int v4u __attribute__((ext_vector_type(4)));
typedef int v8i_t __attribute__((ext_vector_type(8)));
typedef int v4i_t __attribute__((ext_vector_type(4)));

#define UU   100           // hidden size U
#define TT   200           // time steps
#define EE   100           // input feature size
#define NB   4096          // batch
#define N3   300           // 3*U
#define N3P  304           // 3*U padded to 16
#define UP   112           // U padded to 16

#define HAVE_TDM __has_builtin(__builtin_amdgcn_tensor_load_to_lds)

__device__ __forceinline__ float sigmf(float x) { return 1.0f / (1.0f + __expf(-x)); }

// D = A(16x4 f32) * B(4x16 f32) + C, full fp32 matrix pipe on gfx1250
__device__ __forceinline__ v8f wmma4(v2f a, v2f b, v8f c) {
    return __builtin_amdgcn_wmma_f32_16x16x4_f32(
        false, a, false, b, (short)0, c, false, false);
}

#if HAVE_TDM
// TDM 2D tile load: rows x cols f32 tile, row stride in elements, into LDS
// at byte offset lds_off (dynamic LDS starts at LDS offset 0 - no static LDS).
__device__ __forceinline__ void tdm_load_tile(const float* gsrc, unsigned lds_off,
                                              int rows, int cols, long stride_elems)
{
    unsigned long long ga = (unsigned long long)(uintptr_t)gsrc;
    v4u g0;
    g0.x = 1u;                                            // count=1 user descriptor
    g0.y = lds_off;                                       // lds_addr [63:32]
    g0.z = (unsigned)(ga & 0xFFFFFFFFu);                  // global_addr lo
    g0.w = ((unsigned)(ga >> 32) & 0x01FFFFFFu) | 0x80000000u; // addr[56:32] | type=2
    v8i_t g1;
    g1[0] = 0x00020000;                                   // data_size=2 (4B elements)
    g1[1] = (cols & 0xFFFF) << 16;                        // tensor_dim0[15:0] @ bits 63:48
    g1[2] = ((unsigned)cols >> 16) | ((rows & 0xFFFF) << 16); // dim0 hi | tensor_dim1 lo
    g1[3] = ((unsigned)rows >> 16) | ((cols & 0xFFFF) << 16); // dim1 hi | tile_dim0
    g1[4] = rows & 0xFFFF;                                // tile_dim1 | tile_dim2=0
    g1[5] = (int)(stride_elems & 0xFFFFFFFFll);           // tensor_dim0_stride lo
    g1[6] = (int)((stride_elems >> 32) & 0xFFFFll);       // stride hi | dim1_stride lo=0
    g1[7] = 0;
    v4i_t z4 = {0, 0, 0, 0};
    v8i_t z8 = {0, 0, 0, 0, 0, 0, 0, 0};
    __builtin_amdgcn_tensor_load_to_lds(g0, g1, z4, z4, z8, 0);
}
#endif

// Scan-kernel LDS layout (float indices); x tile is double buffered at offset 0
#define SC_XS    0            // 2 * 16*100
#define SC_HS    3200         // 16*100
#define SC_OS    4800         // 16*100
#define SC_MXS   6400         // 16*304
#define SC_MIS   11264        // 16*304
#define SC_MRW   16128        // 16
#define SC_TOT   16144

// ---------------------------------------------------------------------------
// Fused GRU scan: 16 batch rows / block, 19 waves (one per 16-wide N tile of
// the 304-padded 3U dim). B operands (kernel & rec_kernel slabs) live in
// VGPRs for all 200 steps; X tiles arrive by async TDM into double-buffered
// LDS; h/out state carried in LDS.
// ---------------------------------------------------------------------------
extern "C" __global__ __launch_bounds__(608)
void gru_scan_kernel(const float* __restrict__ X, const float* __restrict__ mask,
                     const float* __restrict__ Wk, const float* __restrict__ Wr,
                     const float* __restrict__ bias,
                     float* __restrict__ all_state, float* __restrict__ last)
{
    extern __shared__ float sm[];
    float* x_s  = sm + SC_XS;
    float* h_s  = sm + SC_HS;
    float* o_s  = sm + SC_OS;
    float* mx_s = sm + SC_MXS;
    float* mi_s = sm + SC_MIS;
    float* mrow = sm + SC_MRW;

    const int tid  = threadIdx.x;
    const int b0   = blockIdx.x * 16;
    const int lane = tid & 31;
    const int w    = tid >> 5;              // wave 0..18 -> N tile
    const int m_l  = lane & 15;
    const int kk   = (lane >> 4) * 2;       // K sub-offset (f32 WMMA layout)
    const int n_l  = 16 * w + m_l;          // output column 0..303
    const int r0   = (lane >> 4) * 8;       // C/D row base per lane half

    // Hoist B fragments (one 16-col slab of each weight matrix) into VGPRs.
    v2f Bk[25], Br[25];
    #pragma unroll
    for (int i = 0; i < 25; ++i) {
        int k4 = 4 * i;
        bool ok = n_l < N3;
        Bk[i].x = ok ? Wk[(k4 + kk) * N3 + n_l] : 0.0f;
        Bk[i].y = ok ? Wk[(k4 + kk + 1) * N3 + n_l] : 0.0f;
        Br[i].x = ok ? Wr[(k4 + kk) * N3 + n_l] : 0.0f;
        Br[i].y = ok ? Wr[(k4 + kk + 1) * N3 + n_l] : 0.0f;
    }

    for (int i = tid; i < 16 * UU; i += blockDim.x) { h_s[i] = 0.0f; o_s[i] = 0.0f; }

    // Per-thread gate-phase precompute (element mapping, biases, output bases)
    int em[3], eu[3], nelem = 0;
    float bz0[3], bz1[3], br0f[3], br1f[3], bh0[3], bh1[3];
    size_t obase[3];
    #pragma unroll
    for (int j = 0; j < 3; ++j) {
        int e = tid + 608 * j;
        if (e < 16 * UU) {
            int m = e / UU, u = e % UU;
            em[j] = m; eu[j] = u;
            bz0[j] = bias[u];            bz1[j] = bias[N3 + u];
            br0f[j] = bias[UU + u];      br1f[j] = bias[N3 + UU + u];
            bh0[j] = bias[2 * UU + u];   bh1[j] = bias[N3 + 2 * UU + u];
            obase[j] = (size_t)(b0 + m) * TT * UU + u;
            nelem = j + 1;
        }
    }

#if HAVE_TDM
    if (w == 0)   // kick off tile t=0
        tdm_load_tile(X + (size_t)b0 * TT * EE, SC_XS * 4, 16, EE, (long)TT * EE);
#endif

    for (int t = 0; t < TT; ++t) {
        float* xcur = x_s + (t & 1) * (16 * EE);
#if HAVE_TDM
        if (w == 0) __builtin_amdgcn_s_wait_tensorcnt(0);
#else
        for (int i = tid; i < 16 * EE; i += blockDim.x) {
            int m = i / EE, e = i % EE;
            xcur[i] = X[((size_t)(b0 + m) * TT + t) * EE + e];
        }
#endif
        if (tid < 16) mrow[tid] = mask[(size_t)(b0 + tid) * TT + t];
        __syncthreads();
#if HAVE_TDM
        if (w == 0 && t + 1 < TT)   // async copy of next tile overlaps compute
            tdm_load_tile(X + ((size_t)b0 * TT + (t + 1)) * EE,
                          (SC_XS + ((t + 1) & 1) * 16 * EE) * 4, 16, EE, (long)TT * EE);
#endif

        // mx = x@kernel, mi = h@rec_kernel : 50 fp32 WMMAs, A from LDS, B in VGPRs
        v8f ax = {}; v8f ai = {};
        #pragma unroll
        for (int i = 0; i < 25; ++i) {
            int k4 = 4 * i;
            v2f a  = *(const v2f*)(xcur + m_l * EE + k4 + kk);
            ax = wmma4(a, Bk[i], ax);
            v2f ah = *(const v2f*)(h_s + m_l * UU + k4 + kk);
            ai = wmma4(ah, Br[i], ai);
        }
        #pragma unroll
        for (int r = 0; r < 8; ++r) {
            mx_s[(r + r0) * N3P + n_l] = ax[r];
            mi_s[(r + r0) * N3P + n_l] = ai[r];
        }
        __syncthreads();

        // Gates + state update
        for (int j = 0; j < nelem; ++j) {
            int m = em[j], u = eu[j];
            float xz = mx_s[m * N3P + u] + bz0[j];
            float rz = mi_s[m * N3P + u] + bz1[j];
            float xr = mx_s[m * N3P + UU + u] + br0f[j];
            float rr = mi_s[m * N3P + UU + u] + br1f[j];
            float xh = mx_s[m * N3P + 2 * UU + u] + bh0[j];
            float rh = mi_s[m * N3P + 2 * UU + u] + bh1[j];
            float z  = sigmf(xz + rz);
            float r  = sigmf(xr + rr);
            float hh = tanhf(xh + r * rh);
            float hold = h_s[m * UU + u];
            float hnew = z * hold + (1.0f - z) * hh;
            bool  mk   = mrow[m] > 0.5f;
            float hnx  = mk ? hnew : hold;
            float onx  = mk ? hnew : o_s[m * UU + u];
            h_s[m * UU + u] = hnx;
            o_s[m * UU + u] = onx;
            all_state[obase[j] + (size_t)t * UU] = onx;
        }
        __syncthreads();
    }
    for (int j = 0; j < nelem; ++j)
        last[(size_t)(b0 + em[j]) * UU + eu[j]] = h_s[em[j] * UU + eu[j]];
}

// ---------------------------------------------------------------------------
// p = last_state @ A1  (one 16x16 tile per wave)
// ---------------------------------------------------------------------------
extern "C" __global__ __launch_bounds__(32)
void p_kernel(const float* __restrict__ last, const float* __restrict__ A1,
              float* __restrict__ p)
{
    const int lane = threadIdx.x & 31;
    const int b0   = blockIdx.x * 16;
    const int m_l  = lane & 15;
    const int kk   = (lane >> 4) * 2;
    const int n    = blockIdx.y * 16 + m_l;   // 0..111
    v8f acc = {};
    for (int k4 = 0; k4 < UU; k4 += 4) {
        v2f a = *(const v2f*)(last + (size_t)(b0 + m_l) * UU + k4 + kk);
        v2f b;
        b.x = (n < UU) ? A1[(k4 + kk) * UU + n] : 0.0f;
        b.y = (n < UU) ? A1[(k4 + kk + 1) * UU + n] : 0.0f;
        acc = wmma4(a, b, acc);
    }
    if (n < UU) {
        int r0 = (lane >> 4) * 8;
        #pragma unroll
        for (int r = 0; r < 8; ++r)
            p[(size_t)(b0 + r + r0) * UU + n] = acc[r];
    }
}

// Attention-kernel LDS layout (float indices); S tile double buffered at 0
#define AT_SS    0            // 2 * 16*100
#define AT_PS    3200         // 16*100
#define AT_VS    4800         // 100
#define AT_QS    4900         // 16*112
#define AT_RED   6692         // 16*16
#define AT_ALPHA 6948         // 16
#define AT_MRW   6964         // 16
#define AT_ACC   6980         // 16*100
#define AT_TOT   8580

// ---------------------------------------------------------------------------
// Attention: g = sigmoid(mask*p + all_state@A2); alpha = g.v; out = sum alpha*S
// 16 batch rows / block, 7 waves; A2 slab in VGPRs, S tiles via async TDM.
// ---------------------------------------------------------------------------
extern "C" __global__ __launch_bounds__(224)
void attn_kernel(const float* __restrict__ all_state, const float* __restrict__ mask,
                 const float* __restrict__ A2, const float* __restrict__ v,
                 const float* __restrict__ p, float* __restrict__ out)
{
    extern __shared__ float sm[];
    float* S_s   = sm + AT_SS;
    float* p_s   = sm + AT_PS;
    float* v_s   = sm + AT_VS;
    float* q_s   = sm + AT_QS;
    float* red   = sm + AT_RED;
    float* alpha = sm + AT_ALPHA;
    float* mrow  = sm + AT_MRW;
    float* acc   = sm + AT_ACC;

    const int tid  = threadIdx.x;
    const int b0   = blockIdx.x * 16;
    const int lane = tid & 31;
    const int w    = tid >> 5;          // 0..6
    const int m_l  = lane & 15;
    const int kk   = (lane >> 4) * 2;
    const int n_l  = 16 * w + m_l;      // 0..111
    const int r0   = (lane >> 4) * 8;
    const int gm   = tid & 15;
    const int grp  = tid >> 4;          // 0..13

    // A2 slab in registers
    v2f Ba[25];
    #pragma unroll
    for (int i = 0; i < 25; ++i) {
        int k4 = 4 * i;
        bool ok = n_l < UU;
        Ba[i].x = ok ? A2[(k4 + kk) * UU + n_l] : 0.0f;
        Ba[i].y = ok ? A2[(k4 + kk + 1) * UU + n_l] : 0.0f;
    }
    for (int i = tid; i < 16 * UU; i += blockDim.x) {
        int m = i / UU, u = i % UU;
        p_s[i] = p[(size_t)(b0 + m) * UU + u];
        acc[i] = 0.0f;
    }
    for (int i = tid; i < UU; i += blockDim.x) v_s[i] = v[i];

#if HAVE_TDM
    if (w == 0)
        tdm_load_tile(all_state + (size_t)b0 * TT * UU, AT_SS * 4, 16, UU, (long)TT * UU);
#endif

    for (int t = 0; t < TT; ++t) {
        float* Scur = S_s + (t & 1) * (16 * UU);
#if HAVE_TDM
        if (w == 0) __builtin_amdgcn_s_wait_tensorcnt(0);
#else
        for (int i = tid; i < 16 * UU; i += blockDim.x) {
            int m = i / UU, u = i % UU;
            Scur[i] = all_state[((size_t)(b0 + m) * TT + t) * UU + u];
        }
#endif
        if (tid < 16) mrow[tid] = mask[(size_t)(b0 + tid) * TT + t];
        __syncthreads();
#if HAVE_TDM
        if (w == 0 && t + 1 < TT)
            tdm_load_tile(all_state + ((size_t)b0 * TT + (t + 1)) * UU,
                          (AT_SS + ((t + 1) & 1) * 16 * UU) * 4, 16, UU, (long)TT * UU);
#endif

        // q = S @ A2 via fp32 WMMA
        v8f q = {};
        #pragma unroll
        for (int i = 0; i < 25; ++i) {
            int k4 = 4 * i;
            v2f a = *(const v2f*)(Scur + m_l * UU + k4 + kk);
            q = wmma4(a, Ba[i], q);
        }
        #pragma unroll
        for (int r = 0; r < 8; ++r) q_s[(r + r0) * UP + n_l] = q[r];
        __syncthreads();

        // alpha[m] = sum_u sigmoid(mask*p + q) * v[u]  (14-way partials per row)
        float partial = 0.0f;
        float mk = mrow[gm];
        for (int u = grp; u < UU; u += 14)
            partial += sigmf(mk * p_s[gm * UU + u] + q_s[gm * UP + u]) * v_s[u];
        red[gm * 16 + grp] = partial;
        __syncthreads();
        if (tid < 16) {
            float s = 0.0f;
            for (int j = 0; j < 14; ++j) s += red[tid * 16 + j];
            alpha[tid] = s;
        }
        __syncthreads();
        for (int i = tid; i < 16 * UU; i += blockDim.x)
            acc[i] += alpha[i / UU] * Scur[i];
        __syncthreads();
    }
    for (int i = tid; i < 16 * UU; i += blockDim.x) {
        int m = i / UU, u = i % UU;
        out[(size_t)(b0 + m) * UU + u] = acc[i];
    }
}

extern "C" void kernel_launch(void* const* d_in, const int* in_sizes, int n_in,
                              void* d_out, int out_size, void* d_ws, size_t ws_size,
                              hipStream_t stream)
{
    const float* X    = (const float*)d_in[0];   // [B,T,E]
    const float* mask = (const float*)d_in[1];   // [B,T]
    const float* Wk   = (const float*)d_in[2];   // [E,3U]
    const float* Wr   = (const float*)d_in[3];   // [U,3U]
    const float* bias = (const float*)d_in[4];   // [2,3U]
    const float* A1   = (const float*)d_in[5];   // [U,U]
    const float* A2   = (const float*)d_in[6];   // [U,U]
    const float* v    = (const float*)d_in[7];   // [1,U]
    float* out = (float*)d_out;                  // [B,U]

    float* all_state = (float*)d_ws;                           // B*T*U f32
    float* last      = all_state + (size_t)NB * TT * UU;       // B*U
    float* p         = last + (size_t)NB * UU;                 // B*U

    gru_scan_kernel<<<dim3(NB / 16), dim3(608), SC_TOT * sizeof(float), stream>>>(
        X, mask, Wk, Wr, bias, all_state, last);

    p_kernel<<<dim3(NB / 16, 7), dim3(32), 0, stream>>>(last, A1, p);

    attn_kernel<<<dim3(NB / 16), dim3(224), AT_TOT * sizeof(float), stream>>>(
        all_state, mask, A2, v, p, out);
}